// SparseLinearSuper_23192823398432
// MI455X (gfx1250) — compile-verified
//
#include <hip/hip_runtime.h>
#include <stdint.h>

// ---------------------------------------------------------------------------
// 2:4 structured-sparse linear (x @ (w*mask)^T + bias) for MI455X / gfx1250.
// Main compute: V_SWMMAC_F32_16X16X64_F16 (2:4 sparse A = weights, dense B = x).
// Fully-unrolled K loop with saddr+imm addressing (constant K strides), LDS-
// transposed coalesced stores, vectorized bias add.
// ---------------------------------------------------------------------------

typedef __attribute__((ext_vector_type(16))) _Float16 v16h;  //  8 VGPR: sparse A frag
typedef __attribute__((ext_vector_type(32))) _Float16 v32h;  // 16 VGPR: dense  B frag
typedef __attribute__((ext_vector_type(8)))  float    v8f;   //  8 VGPR: C/D frag
typedef __attribute__((ext_vector_type(4)))  float    v4f;

#define DIN   768
#define DOUT  3072
#define MTOT  12608          // 64 * 197 tokens
#define KT    12             // DIN  / 64  K-tiles
#define NT    192            // DOUT / 16  N-tiles (A rows = output features)
#define MT    788            // MTOT / 16  M-tiles (B cols = tokens)
#define NBLK  48             // DOUT / 64 : waves along N (4 n-tiles per wave)
#define MBLK  394            // MTOT / 32 : waves along M (2 m-tiles per wave)

__device__ __forceinline__ uint32_t pack_f16x2(float lo, float hi) {
  _Float16 hl = (_Float16)lo, hh = (_Float16)hi;
  uint16_t ul = __builtin_bit_cast(uint16_t, hl);
  uint16_t uh = __builtin_bit_cast(uint16_t, hh);
  return (uint32_t)ul | ((uint32_t)uh << 16);
}

// ---------------------------------------------------------------------------
// Kernel 1: compacted 2:4 weight fragments (f16) + 2-bit index words in the
// exact SWMMAC A-fragment lane/VGPR layout (ISA 7.12.2 / 7.12.4).
//   wcf : [NT][KT][32 lanes][8 dwords]  (v16h per lane)
//   widx: [NT][KT][32 lanes]            (one u32 index word per lane)
// ---------------------------------------------------------------------------
__global__ void prep_weights(const float* __restrict__ w,
                             uint32_t* __restrict__ wcf,
                             uint32_t* __restrict__ widx) {
  int t = blockIdx.x * blockDim.x + threadIdx.x;
  if (t >= NT * KT * 32) return;
  int lane = t & 31;
  int kt   = (t >> 5) % KT;
  int nt   = t / (KT * 32);
  int row  = nt * 16 + (lane & 15);   // output feature (A-matrix row M)
  int half = lane >> 4;               // lane half selects K sub-range

  const float* wr = w + (size_t)row * DIN + kt * 64;
  size_t off = (size_t)(nt * KT + kt) * 32 + lane;

  uint32_t idxw = 0;
  float v0[16], v1[16];
  int   k0a[16], k1a[16];

  #pragma unroll
  for (int g = 0; g < 16; ++g) {      // 16 groups of 4 per 64-wide K tile
    float w4[4], a4[4];
    #pragma unroll
    for (int e = 0; e < 4; ++e) { w4[e] = wr[g * 4 + e]; a4[e] = fabsf(w4[e]); }
    // prune the two smallest |w| (first index wins ties, matching topk(-|w|))
    int p0 = 0;
    #pragma unroll
    for (int e = 1; e < 4; ++e) if (a4[e] < a4[p0]) p0 = e;
    int p1 = -1;
    #pragma unroll
    for (int e = 0; e < 4; ++e)
      if (e != p0 && (p1 < 0 || a4[e] < a4[p1])) p1 = e;
    int k0 = -1, k1 = -1;             // kept indices, ascending (idx0 < idx1)
    #pragma unroll
    for (int e = 0; e < 4; ++e)
      if (e != p0 && e != p1) { if (k0 < 0) k0 = e; else k1 = e; }
    k0a[g] = k0; k1a[g] = k1; v0[g] = w4[k0]; v1[g] = w4[k1];

    // index word: lanes 0-15 carry groups 0..7, lanes 16-31 carry groups 8..15
    if ((g >> 3) == half)
      idxw |= (uint32_t)(k0 | (k1 << 2)) << (4 * (g & 7));
  }
  widx[off] = idxw;

  // value fragments: compacted col c = 2g; this lane stores c' = c - half*8
  // when c' in {0,2,4,6} (VGPR 0-3) or {16,18,20,22} (VGPR 4-7).
  #pragma unroll
  for (int g = 0; g < 16; ++g) {
    int cc = 2 * g - half * 8;
    bool mine = (cc >= 0) && ((cc < 8) || (cc >= 16 && cc < 24));
    if (mine) {
      int r = (cc < 8) ? (cc >> 1) : 4 + ((cc - 16) >> 1);
      wcf[off * 8 + r] = pack_f16x2(v0[g], v1[g]);
    }
  }
}

// ---------------------------------------------------------------------------
// Kernel 2: fp32 x -> f16 B fragments (64x16 column layout, ISA 7.12.4):
//   lanes 0-15 : K 0..15 (V0-7) and 32..47 (V8-15), column N = lane
//   lanes 16-31: K 16..31        and 48..63,        column N = lane-16
//   xf : [MT][KT][32 lanes][16 dwords]  (v32h per lane)
// ---------------------------------------------------------------------------
__global__ void prep_x(const float* __restrict__ x, uint32_t* __restrict__ xf) {
  int t = blockIdx.x * blockDim.x + threadIdx.x;
  if (t >= MT * KT * 32) return;
  int lane = t & 31;
  int kt   = (t >> 5) % KT;
  int mt   = t / (KT * 32);
  int m    = mt * 16 + (lane & 15);   // token (B-matrix column N)
  int koff = (lane >> 4) * 16;
  const float* xr = x + (size_t)m * DIN + kt * 64;
  size_t off = ((size_t)(mt * KT + kt) * 32 + lane) * 16;
  #pragma unroll
  for (int v = 0; v < 16; ++v) {
    int K = koff + ((v < 8) ? 2 * v : 32 + 2 * (v - 8));
    xf[off + v] = pack_f16x2(xr[K], xr[K + 1]);
  }
}

// ---------------------------------------------------------------------------
// Kernel 3: sparse GEMM. One wave -> 32 tokens x 64 outputs (2x4 fragments).
// Fully-unrolled K loop; all fragment streams have constant K strides, so
// every load is saddr + 32-bit lane offset + immediate (computed once):
//   A frag : +1024 B / kt     idx word : +128 B / kt     B frag : +2048 B / kt
// Scheduler hoists next-step loads over the 8 swmmacs of the current step.
// ---------------------------------------------------------------------------
__global__ void __launch_bounds__(256)
sparse_gemm(const uint32_t* __restrict__ wcf, const uint32_t* __restrict__ widx,
            const uint32_t* __restrict__ xf,  const float* __restrict__ bias,
            float* __restrict__ out) {
  __shared__ __align__(16) float lds_tile[8 * 32 * 64];   // 64 KB / block

  int gw   = (int)((blockIdx.x * blockDim.x + threadIdx.x) >> 5);
  int lane = threadIdx.x & 31;
  int nb   = gw % NBLK;           // 64-output block
  int mb   = gw / NBLK;           // 32-token block
  if (mb >= MBLK) return;         // wave-uniform (grid is exact; never taken)

  v8f c[2][4];
  #pragma unroll
  for (int i = 0; i < 2; ++i)
    #pragma unroll
    for (int j = 0; j < 4; ++j) c[i][j] = {};

  // Per-lane 32-bit byte offsets, computed once.
  const char* wcp = (const char*)wcf;
  const char* wip = (const char*)widx;
  const char* xfp = (const char*)xf;
  uint32_t abase[4], ibase[4], bbase[2];
  #pragma unroll
  for (int j = 0; j < 4; ++j) {
    uint32_t slot = (uint32_t)((nb * 4 + j) * KT) * 32u + (uint32_t)lane;
    abase[j] = slot * 32u;                 // v16h = 32 B per lane
    ibase[j] = slot * 4u;                  // u32 index word per lane
  }
  #pragma unroll
  for (int i = 0; i < 2; ++i)
    bbase[i] = ((uint32_t)((mb * 2 + i) * KT) * 32u + (uint32_t)lane) * 64u;  // v32h

  #pragma unroll
  for (int kt = 0; kt < KT; ++kt) {
    v16h a[4]; uint32_t ix[4]; v32h b[2];
    #pragma unroll
    for (int j = 0; j < 4; ++j) {
      a[j]  = *(const v16h*)(const void*)(wcp + abase[j] + kt * 1024);
      ix[j] = *(const uint32_t*)(const void*)(wip + ibase[j] + kt * 128);
    }
    #pragma unroll
    for (int i = 0; i < 2; ++i)
      b[i] = *(const v32h*)(const void*)(xfp + bbase[i] + kt * 2048);

    #pragma unroll
    for (int i = 0; i < 2; ++i)
      #pragma unroll
      for (int j = 0; j < 4; ++j)
        // D = sparse(A_weights) x B_x + C   (2:4, 2x dense WMMA rate)
        c[i][j] = __builtin_amdgcn_swmmac_f32_16x16x64_f16(
            /*neg_a=*/false, a[j], /*neg_b=*/false, b[i],
            c[i][j], (int)ix[j], /*reuse_a=*/false, /*reuse_b=*/false);
  }

  // ---- Epilogue ----
  // D frag element (vgpr v, lane): output row o = j*16 + vrow + v (contiguous
  // in v), token = lane%16 (lane and lane+16 share a token).
  int mlane = lane & 15;
  int half  = lane >> 4;
  int vrow  = half ? 8 : 0;
  float* lds = &lds_tile[(threadIdx.x >> 5) * (32 * 64)];

  #pragma unroll
  for (int i = 0; i < 2; ++i) {
    int tok_local = i * 16 + mlane;
    #pragma unroll
    for (int j = 0; j < 4; ++j) {
      v8f bv = *(const v8f*)(bias + (nb * 4 + j) * 16 + vrow);  // 32B-aligned
      v8f d  = c[i][j] + bv;
      *(v8f*)(&lds[tok_local * 64 + j * 16 + vrow]) = d;        // 2x ds_store_b128
    }
  }
  // wave-private LDS slice: same-wave RAW ordering enforced via DScnt.

  // Coalesced write-out: per iteration, lanes 0-15 cover token 2r, lanes
  // 16-31 cover token 2r+1 -> 16 global_store_b128 per wave, immediate
  // offsets off a single address register.
  int chunk = lane & 15;
  const float* ldsr = &lds[half * 64 + chunk * 4];
  float* outp = out + ((size_t)(mb * 32 + half)) * DOUT + nb * 64 + chunk * 4;
  #pragma unroll
  for (int r = 0; r < 16; ++r) {
    v4f d = *(const v4f*)(ldsr + r * 2 * 64);
    *(v4f*)(outp + (size_t)r * 2 * DOUT) = d;
  }
}

// ---------------------------------------------------------------------------
// Host side
// ---------------------------------------------------------------------------
extern "C" void kernel_launch(void* const* d_in, const int* in_sizes, int n_in,
                              void* d_out, int out_size, void* d_ws, size_t ws_size,
                              hipStream_t stream) {
  const float* x    = (const float*)d_in[0];   // [64,197,768]
  const float* w    = (const float*)d_in[1];   // [3072,768]
  const float* bias = (const float*)d_in[2];   // [3072]
  float*       out  = (float*)d_out;           // [64,197,3072]

  char* ws = (char*)d_ws;
  const size_t WCF_BYTES  = (size_t)NT * KT * 32 * 8 * 4;   // 2,359,296
  const size_t WIDX_BYTES = (size_t)NT * KT * 32 * 4;       //   294,912
  uint32_t* wcf  = (uint32_t*)(ws);
  uint32_t* widx = (uint32_t*)(ws + WCF_BYTES);
  uint32_t* xf   = (uint32_t*)(ws + WCF_BYTES + WIDX_BYTES);

  prep_weights<<<(NT * KT * 32) / 256, 256, 0, stream>>>(w, wcf, widx);
  prep_x      <<<(MT * KT * 32) / 256, 256, 0, stream>>>(x, xf);

  // 394 * 48 = 18912 waves, 8 waves / block -> 2364 blocks
  sparse_gemm <<<(MBLK * NBLK) / 8, 256, 0, stream>>>(wcf, widx, xf, bias, out);
}